// GraphIdentifierEnvNonlinear_55147380080687
// MI455X (gfx1250) — compile-verified
//
#include <hip/hip_runtime.h>

typedef float v2f __attribute__((ext_vector_type(2)));
typedef float v8f __attribute__((ext_vector_type(8)));

#define NN 1024
#define BB 4

#define LOG2E 1.4426950408889634f

// ---- branch-free tanh: gfx1250 V_TANH_F32 (probe-confirmed last round) ----
#if defined(__has_builtin)
#  if __has_builtin(__builtin_amdgcn_tanhf)
#    define FAST_TANH(x) __builtin_amdgcn_tanhf(x)
#  elif __has_builtin(__builtin_amdgcn_tanh_f32)
#    define FAST_TANH(x) __builtin_amdgcn_tanh_f32(x)
#  endif
#endif
#ifndef FAST_TANH
__device__ __forceinline__ float fast_tanh_impl(float x) {
  float xc = fminf(fmaxf(x, -10.f), 10.f);
  float e  = __builtin_amdgcn_exp2f(xc * (2.0f * LOG2E));
  return 1.0f - 2.0f * __builtin_amdgcn_rcpf(e + 1.0f);
}
#  define FAST_TANH(x) fast_tanh_impl(x)
#endif

// Kernel 1: A[i,j] = softmax(Theta[i,:]) masked-diag + renormalized.
// softmax -> mask -> renorm == exp(theta - max)*(j!=i) / sum_{j!=i} exp(theta - max)
__global__ __launch_bounds__(256) void row_softmax_kernel(const float* __restrict__ Theta,
                                                          float* __restrict__ A) {
  const int i = blockIdx.x;
  const int t = threadIdx.x;
  __shared__ float red[256];

  float m = -3.4e38f;
  for (int j = t; j < NN; j += 256) m = fmaxf(m, Theta[i * NN + j]);
  red[t] = m;
  __syncthreads();
  for (int s = 128; s > 0; s >>= 1) {
    if (t < s) red[t] = fmaxf(red[t], red[t + s]);
    __syncthreads();
  }
  m = red[0];
  __syncthreads();

  float sum = 0.f;
  for (int j = t; j < NN; j += 256) {
    float e = __builtin_amdgcn_exp2f((Theta[i * NN + j] - m) * LOG2E);
    if (j != i) sum += e;
  }
  red[t] = sum;
  __syncthreads();
  for (int s = 128; s > 0; s >>= 1) {
    if (t < s) red[t] += red[t + s];
    __syncthreads();
  }
  const float inv = 1.0f / red[0];

  for (int j = t; j < NN; j += 256) {
    float v = (j == i) ? 0.f
                       : __builtin_amdgcn_exp2f((Theta[i * NN + j] - m) * LOG2E) * inv;
    A[i * NN + j] = v;
  }
}

// Kernel 2: one wave per (b, i). Per 16 j's, one V_WMMA_F32_16X16X4_F32 computes the
// TRANSPOSED hidden layer  D'[h, p] = (W1b^T @ feats^T)[h, p]:
//   A operand = [W1; b1]^T (16x4, constant), B operand = feats^T (4x16, per-j-tile).
// Result layout: hidden index h = r + 8*half lives on the lane's 8 registers,
// pair index p = lane%16 lives on lanes  ->  layer-2 contraction over h is a pure
// per-lane register dot product + ONE shfl_xor(16) to merge the two K-halves.
__global__ __launch_bounds__(256) void fused_kernel(const float* __restrict__ x,
                                                    const float* __restrict__ A,
                                                    const float* __restrict__ W1,
                                                    const float* __restrict__ b1,
                                                    const float* __restrict__ W2,
                                                    const float* __restrict__ b2,
                                                    float* __restrict__ out) {
  const int lane = threadIdx.x & 31;
  const int wid  = (blockIdx.x * blockDim.x + threadIdx.x) >> 5;
  const int b    = wid >> 10;          // wid / N
  const int i    = wid & (NN - 1);     // wid % N
  const int half = lane >> 4;          // K=0,1 in lanes 0-15; K=2,3 in lanes 16-31
  const int l16  = lane & 15;

  const float* __restrict__ xb   = x + b * NN;
  const float* __restrict__ Arow = A + i * NN;
  const float xi = xb[i];

  // A operand (16x4, constant across the loop): rows = hidden h, cols = K feats.
  // W1b[k][h] with k: 0=xi-weight, 1=xj-weight, 2=diff-weight, 3=bias row (b1).
  v2f Wm;
  Wm.x = (half == 0) ? W1[0 * 16 + l16] : W1[2 * 16 + l16];
  Wm.y = (half == 0) ? W1[1 * 16 + l16] : b1[l16];

  // Per-lane slice of W2 for the register-space contraction: h = r + 8*half.
  float w2v[8];
#pragma unroll
  for (int r = 0; r < 8; ++r) w2v[r] = W2[8 * half + r];
  const float bias2 = b2[0];

  float acc = 0.f;
  for (int j0 = 0; j0 < NN; j0 += 16) {
    const float xj   = xb[j0 + l16];     // coalesced (16 consecutive, mirrored halves)
    const float aij  = Arow[j0 + l16];   // coalesced; A[i,i]==0 masks the diagonal
    const float diff = xj - xi;

    // B operand = feats^T (4x16): column p = l16, K split across halves.
    v2f Fm;
    Fm.x = (half == 0) ? xi : diff;
    Fm.y = (half == 0) ? xj : 1.0f;

    v8f C = {};
    v8f D = __builtin_amdgcn_wmma_f32_16x16x4_f32(
        /*neg_a=*/false, Wm, /*neg_b=*/false, Fm,
        /*c_mod=*/(short)0, C, /*reuse_a=*/false, /*reuse_b=*/false);

    // D[r] = pre-activation of hidden unit h = r + 8*half for pair p = l16.
    float partial = 0.f;
#pragma unroll
    for (int r = 0; r < 8; ++r) partial += FAST_TANH(D[r]) * w2v[r];

    // Merge the two hidden halves (h=0..7 from lanes<16, h=8..15 from lanes>=16).
    const float raw = partial + __shfl_xor(partial, 16, 32) + bias2;

    // alpha = 0.5 + sigmoid(raw): v_exp_f32 + v_rcp_f32, branch-free.
    const float e     = __builtin_amdgcn_exp2f(raw * -LOG2E);
    const float alpha = 0.5f + __builtin_amdgcn_rcpf(1.0f + e);

    acc += aij * alpha * diff;
  }

  // Sum over the 16 pairs-per-tile lanes (halves hold identical copies).
#pragma unroll
  for (int m = 1; m <= 8; m <<= 1) acc += __shfl_xor(acc, m, 32);
  if (lane == 0) out[b * NN + i] = xi + 0.1f * acc;
}

extern "C" void kernel_launch(void* const* d_in, const int* in_sizes, int n_in,
                              void* d_out, int out_size, void* d_ws, size_t ws_size,
                              hipStream_t stream) {
  (void)in_sizes; (void)n_in; (void)out_size; (void)ws_size;
  const float* x     = (const float*)d_in[0];
  const float* Theta = (const float*)d_in[1];
  const float* W1    = (const float*)d_in[2];
  const float* b1    = (const float*)d_in[3];
  const float* W2    = (const float*)d_in[4];
  const float* b2    = (const float*)d_in[5];
  float* out = (float*)d_out;
  float* A   = (float*)d_ws;   // NN*NN floats = 4 MiB scratch

  row_softmax_kernel<<<NN, 256, 0, stream>>>(Theta, A);

  const int waves   = BB * NN;              // one wave per (b, i)
  const int threads = 256;                  // 8 waves per block
  const int blocks  = waves * 32 / threads; // 512
  fused_kernel<<<blocks, threads, 0, stream>>>(x, A, W1, b1, W2, b2, out);
}